// LightGCNCondensed_72645076845131
// MI455X (gfx1250) — compile-verified
//
#include <hip/hip_runtime.h>
#include <hip/hip_bf16.h>
#include <math.h>

// LightGCN condensed propagation for MI455X (gfx1250, wave32).
//
// CSR-once (counts -> scan -> fill-with-normalization), then atomic-free
// pull-mode propagation: one wave32 per destination row, per-wave async
// (ASYNCcnt) LDS staging of the row's edge list, coalesced 256B random row
// gathers with prefetch lookahead, register accumulation, fused
// next-layer store + running-average update. ~110 MB working set is
// L2-resident (192 MB L2). Hot-loop atomics: zero.

#define DIM       64
#define N_LAYERS  3
#define EPSF      1e-8f
#define BLK       256      // 8 wave32s per block
#define CHUNK     32       // edges staged per wave per async round
#define PF        4        // gather prefetch lookahead (edges)
#define SCAN_T    1024
#define SCAN_I    8        // counts reduced serially per scan thread

__device__ __forceinline__ void atomic_add_f32(float* p, float v) {
  __hip_atomic_fetch_add(p, v, __ATOMIC_RELAXED, __HIP_MEMORY_SCOPE_AGENT);
}
__device__ __forceinline__ void atomic_inc_u32(int* p) {
  __hip_atomic_fetch_add(p, 1, __ATOMIC_RELAXED, __HIP_MEMORY_SCOPE_AGENT);
}
__device__ __forceinline__ int atomic_add_ret_u32(int* p) {
  return __hip_atomic_fetch_add(p, 1, __ATOMIC_RELAXED, __HIP_MEMORY_SCOPE_AGENT);
}

// Async DMA: 4 bytes per lane, global -> LDS (tracked by ASYNCcnt).
__device__ __forceinline__ void async_load_b32(void* lds_dst, const void* gsrc) {
  unsigned           ldsa  = (unsigned)(size_t)lds_dst;   // low 32 bits = LDS offset
  unsigned long long gaddr = (unsigned long long)(size_t)gsrc;
  asm volatile("global_load_async_to_lds_b32 %0, %1, off"
               : : "v"(ldsa), "v"(gaddr) : "memory");
}
__device__ __forceinline__ void wait_asynccnt0() {
  asm volatile("s_wait_asynccnt 0" ::: "memory");
}

// ---------------------------------------------------------------------------
// 1) softplus weights + float degree sums + integer degree counts.
// ---------------------------------------------------------------------------
__global__ __launch_bounds__(BLK) void k_edge_w_deg(
    const float* __restrict__ logit,
    const int*   __restrict__ cu, const int* __restrict__ ci,
    float* __restrict__ w,
    float* __restrict__ deg_u, float* __restrict__ deg_i,
    int*   __restrict__ cnt_u, int*   __restrict__ cnt_i,
    int nE) {
  int e = blockIdx.x * blockDim.x + threadIdx.x;
  if (e >= nE) return;
  float x  = logit[e];
  float sp = (x > 0.f) ? (x + log1pf(__expf(-x))) : log1pf(__expf(x));
  float wv = sp + EPSF;
  w[e] = wv;
  int u = cu[e], it = ci[e];
  atomic_add_f32(deg_u + u,  wv);
  atomic_add_f32(deg_i + it, wv);
  atomic_inc_u32(cnt_u + u);
  atomic_inc_u32(cnt_i + it);
}

// ---------------------------------------------------------------------------
// 2) Exclusive scan of counts -> row offsets (+ cursor copy).
//    Single workgroup; each thread serially reduces SCAN_I contiguous counts,
//    1024 partials scanned in LDS -> 8192 counts per chunk iteration.
// ---------------------------------------------------------------------------
__global__ __launch_bounds__(SCAN_T) void k_scan(
    const int* __restrict__ cnt, int* __restrict__ off,
    int* __restrict__ cursor, int n) {
  __shared__ int s[SCAN_T];
  __shared__ int carry;
  const int tid = threadIdx.x;
  if (tid == 0) carry = 0;
  __syncthreads();
  const int CH = SCAN_T * SCAN_I;
  for (int base = 0; base < n; base += CH) {
    int i0 = base + tid * SCAN_I;
    int v[SCAN_I];
    int sum = 0;
#pragma unroll
    for (int j = 0; j < SCAN_I; ++j) {
      int i = i0 + j;
      v[j] = (i < n) ? cnt[i] : 0;
      sum += v[j];
    }
    s[tid] = sum;
    __syncthreads();
    for (int d = 1; d < SCAN_T; d <<= 1) {
      int t = (tid >= d) ? s[tid - d] : 0;
      __syncthreads();
      s[tid] += t;
      __syncthreads();
    }
    int run = s[tid] - sum + carry;
#pragma unroll
    for (int j = 0; j < SCAN_I; ++j) {
      int i = i0 + j;
      if (i < n) { off[i] = run; cursor[i] = run; }
      run += v[j];
    }
    __syncthreads();
    if (tid == 0) carry += s[SCAN_T - 1];
    __syncthreads();
  }
  if (tid == 0) off[n] = carry;
}

// ---------------------------------------------------------------------------
// 3) Scatter-fill CSR for both directions, normalization fused in:
//    csr_w = w / (sqrt(deg_u+eps) * sqrt(deg_i+eps))
// ---------------------------------------------------------------------------
__global__ __launch_bounds__(BLK) void k_fill(
    const int* __restrict__ cu, const int* __restrict__ ci,
    const float* __restrict__ w,
    const float* __restrict__ deg_u, const float* __restrict__ deg_i,
    int* __restrict__ curU, int* __restrict__ curI,
    int* __restrict__ csrU_idx, float* __restrict__ csrU_w,
    int* __restrict__ csrI_idx, float* __restrict__ csrI_w, int nE) {
  int e = blockIdx.x * blockDim.x + threadIdx.x;
  if (e >= nE) return;
  int u = cu[e], it = ci[e];
  float nm = w[e] / (sqrtf(deg_u[u] + EPSF) * sqrtf(deg_i[it] + EPSF));
  int p = atomic_add_ret_u32(curU + u);
  csrU_idx[p] = it; csrU_w[p] = nm;
  int q = atomic_add_ret_u32(curI + it);
  csrI_idx[q] = u;  csrI_w[q] = nm;
}

// ---------------------------------------------------------------------------
// 4) Initialize current embeddings (w + delta) and the accumulator (d_out).
// ---------------------------------------------------------------------------
__global__ __launch_bounds__(BLK) void k_init_embed(
    const float4* __restrict__ uw, const float4* __restrict__ ud,
    const float4* __restrict__ iw, const float4* __restrict__ idl,
    float4* __restrict__ cur, float4* __restrict__ acc,
    int nU4, int nTot4) {
  int i = blockIdx.x * blockDim.x + threadIdx.x;
  if (i >= nTot4) return;
  float4 a, b;
  if (i < nU4) { a = uw[i];       b = ud[i]; }
  else         { a = iw[i - nU4]; b = idl[i - nU4]; }
  float4 v;
  v.x = a.x + b.x; v.y = a.y + b.y; v.z = a.z + b.z; v.w = a.w + b.w;
  cur[i] = v;
  acc[i] = v;
}

// ---------------------------------------------------------------------------
// 5) Pull-mode propagation, both directions, accumulator update fused.
//    One wave32 per destination row; per-wave async staging of the row's
//    (idx, w) list into a private LDS slot; 256B coalesced row gathers with
//    prefetch lookahead; zero atomics; fused acc = (acc + v) * scale.
//    WRITE_NEXT=false on the final layer (nxt is dead there).
// ---------------------------------------------------------------------------
template <bool WRITE_NEXT>
__global__ __launch_bounds__(BLK) void k_row_prop(
    const float* __restrict__ cur_u, const float* __restrict__ cur_i,
    float*       __restrict__ nxt,                 // [user rows][item rows]
    float*       __restrict__ acc,                 // d_out, same layout
    float scale,
    const int* __restrict__ offU, const int* __restrict__ csrU_idx,
    const float* __restrict__ csrU_w,
    const int* __restrict__ offI, const int* __restrict__ csrI_idx,
    const float* __restrict__ csrI_w,
    int nU, int nRows) {
  __shared__ int   s_idx[BLK / 32][CHUNK];
  __shared__ float s_w  [BLK / 32][CHUNK];
  const int lane = threadIdx.x & 31;     // wave32
  const int wv   = threadIdx.x >> 5;

  int row = blockIdx.x * (BLK / 32) + wv;
  const int rstride = gridDim.x * (BLK / 32);

  for (; row < nRows; row += rstride) {
    const bool   isU  = (row < nU);
    const int    r    = isU ? row : row - nU;
    const int*   offp = isU ? offU : offI;
    const int*   cidx = isU ? csrU_idx : csrI_idx;
    const float* cw   = isU ? csrU_w : csrI_w;
    const float* src  = isU ? cur_i : cur_u;   // gather from the other side

    const int b0 = offp[r], b1 = offp[r + 1];
    float ax = 0.f, ay = 0.f;

    for (int c = b0; c < b1; c += CHUNK) {
      // per-wave async DMA of this chunk's edge list into our LDS slot
      async_load_b32(&s_idx[wv][lane], cidx + c + lane);
      async_load_b32(&s_w[wv][lane],   cw   + c + lane);
      wait_asynccnt0();   // ASYNCcnt is per-wave: no block barrier needed

      int m = b1 - c;
      if (m > CHUNK) m = CHUNK;
#pragma unroll 4
      for (int k = 0; k < m; ++k) {
        if (k + PF < m) {  // warm the upcoming gather row
          __builtin_prefetch(src + (size_t)s_idx[wv][k + PF] * DIM + lane * 2, 0, 1);
        }
        int   g  = s_idx[wv][k];
        float wg = s_w[wv][k];
        float2 vs = *(const float2*)(src + (size_t)g * DIM + lane * 2);
        ax += vs.x * wg;
        ay += vs.y * wg;
      }
    }

    const size_t o = (size_t)row * DIM + lane * 2;
    if (WRITE_NEXT) {                      // input to the next layer
      float2 nv; nv.x = ax; nv.y = ay;
      *(float2*)(nxt + o) = nv;
    }
    float2 a = *(const float2*)(acc + o);  // fused running-average update
    a.x = (a.x + ax) * scale;
    a.y = (a.y + ay) * scale;
    *(float2*)(acc + o) = a;
  }
}

// ---------------------------------------------------------------------------
extern "C" void kernel_launch(void* const* d_in, const int* in_sizes, int n_in,
                              void* d_out, int out_size, void* d_ws, size_t ws_size,
                              hipStream_t stream) {
  const float* user_w     = (const float*)d_in[0];
  const float* item_w     = (const float*)d_in[1];
  const float* user_delta = (const float*)d_in[2];
  const float* item_delta = (const float*)d_in[3];
  const float* edge_logit = (const float*)d_in[4];
  const int*   cu         = (const int*)d_in[5];
  const int*   ci         = (const int*)d_in[6];
  float*       out        = (float*)d_out;

  const int nE = in_sizes[4];
  const int nU = in_sizes[0] / DIM;
  const int nI = in_sizes[1] / DIM;
  const size_t embU  = (size_t)nU * DIM;
  const size_t emb   = embU + (size_t)nI * DIM;
  const int    nRows = nU + nI;

  // ---- workspace bump allocator (4-byte units, 4-element aligned) ----
  size_t cur4 = 0;
  auto alloc = [&](size_t nelem) {
    size_t p = cur4;
    cur4 += (nelem + 3) & ~(size_t)3;
    return (char*)d_ws + p * 4;
  };
  float* wbuf     = (float*)alloc(nE);                 // softplus weights
  float* deg_u    = (float*)alloc(2 * (size_t)nRows);  // deg_u|deg_i|cnt_u|cnt_i
  float* deg_i    = deg_u + nU;                        // (one contiguous memset)
  int*   cnt_u    = (int*)(deg_i + nI);
  int*   cnt_i    = cnt_u + nU;
  int*   off_u    = (int*)alloc(nU + 1);
  int*   off_i    = (int*)alloc(nI + 1);
  int*   cursor_u = (int*)alloc(nU);
  int*   cursor_i = (int*)alloc(nI);
  int*   csrU_idx = (int*)alloc((size_t)nE + CHUNK);   // +CHUNK pad: unguarded
  float* csrU_w   = (float*)alloc((size_t)nE + CHUNK); //  async staging never
  int*   csrI_idx = (int*)alloc((size_t)nE + CHUNK);   //  reads out of bounds
  float* csrI_w   = (float*)alloc((size_t)nE + CHUNK);
  float* bufA     = (float*)alloc(emb);
  float* bufB     = (float*)alloc(emb);

  // ---- graph preprocessing (once; reused by 3 propagation passes) ----
  hipMemsetAsync(deg_u, 0, 2 * (size_t)nRows * sizeof(float), stream);
  k_edge_w_deg<<<(nE + BLK - 1) / BLK, BLK, 0, stream>>>(
      edge_logit, cu, ci, wbuf, deg_u, deg_i, cnt_u, cnt_i, nE);
  k_scan<<<1, SCAN_T, 0, stream>>>(cnt_u, off_u, cursor_u, nU);
  k_scan<<<1, SCAN_T, 0, stream>>>(cnt_i, off_i, cursor_i, nI);
  k_fill<<<(nE + BLK - 1) / BLK, BLK, 0, stream>>>(
      cu, ci, wbuf, deg_u, deg_i, cursor_u, cursor_i,
      csrU_idx, csrU_w, csrI_idx, csrI_w, nE);

  // ---- embeddings + accumulator ----
  const int nTot4 = (int)(emb / 4);
  const int nU4   = (int)(embU / 4);
  k_init_embed<<<(nTot4 + BLK - 1) / BLK, BLK, 0, stream>>>(
      (const float4*)user_w, (const float4*)user_delta,
      (const float4*)item_w, (const float4*)item_delta,
      (float4*)bufA, (float4*)out, nU4, nTot4);

  // ---- 3 propagation layers, ping-pong bufA/bufB, accum fused ----
  float* curp = bufA;
  float* nxtp = bufB;
  int rowBlocks = (nRows + (BLK / 32) - 1) / (BLK / 32);
  if (rowBlocks > 8192) rowBlocks = 8192;
  for (int l = 0; l < N_LAYERS; ++l) {
    const bool  last = (l == N_LAYERS - 1);
    const float s    = last ? (1.0f / (N_LAYERS + 1)) : 1.0f;
    if (last) {
      k_row_prop<false><<<rowBlocks, BLK, 0, stream>>>(
          curp, curp + embU, nxtp, out, s,
          off_u, csrU_idx, csrU_w, off_i, csrI_idx, csrI_w, nU, nRows);
    } else {
      k_row_prop<true><<<rowBlocks, BLK, 0, stream>>>(
          curp, curp + embU, nxtp, out, s,
          off_u, csrU_idx, csrU_w, off_i, csrI_idx, csrI_w, nU, nRows);
    }
    float* t = curp; curp = nxtp; nxtp = t;
  }
}